// MultiBoxLoss_58729382806031
// MI455X (gfx1250) — compile-verified
//
#include <hip/hip_runtime.h>
#include <math.h>

// ---------------- problem constants (from reference) ----------------
#define B_N   16
#define A_N   65536
#define O_N   24
#define K_TOP 10
#define NSEL  (O_N * 2 * K_TOP)   // 480 selections per batch
#define TPB   1024
#define PER_T (A_N / TPB)         // 64 anchors per thread

// P_TH=0.35, N_TH=0.4, NEGPOS=7, V0=0.1, V1=0.2

// ---------------- vector types ----------------
typedef float v2f __attribute__((ext_vector_type(2)));
typedef float v8f __attribute__((ext_vector_type(8)));
typedef unsigned int v4u __attribute__((ext_vector_type(4)));
typedef int v4i __attribute__((ext_vector_type(4)));
typedef int v8i __attribute__((ext_vector_type(8)));

// ---------------- CDNA5 WMMA wave-sum (f32 16x16x4) ----------------
// D = A(16x4) * ones(4x16) + 0 -> D[M][N] = rowsum(A)[M] for every N.
// total = sum_M D[M][0] = (sum of 8 acc regs at lane 0) + (same at lane 16).
// EXEC must be all ones at the call site.
__device__ __forceinline__ float wave_sum32(float x) {
  v2f a; a[0] = x; a[1] = 0.0f;
  v2f one; one[0] = 1.0f; one[1] = 1.0f;
  v8f c = {};
  c = __builtin_amdgcn_wmma_f32_16x16x4_f32(false, a, false, one,
                                            (short)0, c, false, false);
  float s = c[0]+c[1]+c[2]+c[3]+c[4]+c[5]+c[6]+c[7];
  return __shfl(s, 0, 32) + __shfl(s, 16, 32);
}

// ---------------- helpers ----------------
__device__ __forceinline__ float iou_box(float ax0,float ay0,float ax1,float ay1,
                                         float bx0,float by0,float bx1,float by1){
  float ix = fminf(ax1,bx1) - fmaxf(ax0,bx0);
  float iy = fminf(ay1,by1) - fmaxf(ay0,by0);
  ix = fmaxf(ix, 0.0f); iy = fmaxf(iy, 0.0f);
  float inter = ix * iy;
  float aa = (ax1-ax0)*(ay1-ay0);
  float ab = (bx1-bx0)*(by1-by0);
  return inter / (aa + ab - inter);
}

__device__ __forceinline__ float smooth_l1(float d){
  float ad = fabsf(d);
  return (ad < 1.0f) ? (0.5f * d * d) : (ad - 0.5f);
}

__device__ __forceinline__ unsigned bits2key(unsigned u){  // monotone map
  return (u & 0x80000000u) ? ~u : (u | 0x80000000u);
}
__device__ __forceinline__ float key2f(unsigned k){
  unsigned u = (k & 0x80000000u) ? (k & 0x7FFFFFFFu) : ~k;
  return __uint_as_float(u);
}

// wave32 shfl_xor butterfly + one 32-slot LDS hop: block argmax, tie->min idx
__device__ __forceinline__ void block_argmax(float v, unsigned idx,
                                             float* redv, unsigned* redi, int t,
                                             float& outv, unsigned& outi) {
  #pragma unroll
  for (int off = 16; off > 0; off >>= 1) {
    float ov = __shfl_xor(v, off, 32);
    unsigned oi = __shfl_xor(idx, off, 32);
    if (ov > v || (ov == v && oi < idx)) { v = ov; idx = oi; }
  }
  if ((t & 31) == 0) { redv[t >> 5] = v; redi[t >> 5] = idx; }
  __syncthreads();
  if (t < 32) {
    float v2 = redv[t]; unsigned i2 = redi[t];
    #pragma unroll
    for (int off = 16; off > 0; off >>= 1) {
      float ov = __shfl_xor(v2, off, 32);
      unsigned oi = __shfl_xor(i2, off, 32);
      if (ov > v2 || (ov == v2 && oi < i2)) { v2 = ov; i2 = oi; }
    }
    if (t == 0) { redv[0] = v2; redi[0] = i2; }
  }
  __syncthreads();
  outv = redv[0]; outi = redi[0];
  __syncthreads();                 // protect red* reuse by next call
}

__device__ __forceinline__ int block_sum_i(int v, int* red, int t) {
  #pragma unroll
  for (int off = 16; off > 0; off >>= 1) v += __shfl_xor(v, off, 32);
  if ((t & 31) == 0) red[t >> 5] = v;
  __syncthreads();
  if (t < 32) {
    int x = red[t];
    #pragma unroll
    for (int off = 16; off > 0; off >>= 1) x += __shfl_xor(x, off, 32);
    if (t == 0) red[0] = x;
  }
  __syncthreads();
  int r = red[0];
  __syncthreads();
  return r;
}

__device__ __forceinline__ float block_sum_f(float v, float* red, int t) {
  #pragma unroll
  for (int off = 16; off > 0; off >>= 1) v += __shfl_xor(v, off, 32);
  if ((t & 31) == 0) red[t >> 5] = v;
  __syncthreads();
  if (t < 32) {
    float x = red[t];
    #pragma unroll
    for (int off = 16; off > 0; off >>= 1) x += __shfl_xor(x, off, 32);
    if (t == 0) red[0] = x;
  }
  __syncthreads();
  float r = red[0];
  __syncthreads();
  return r;
}

// ---------------- dynamic-LDS budgets (MI455X: 320KB/WGP) ----------------
static constexpr unsigned SM_ROW  = A_N * 4;        // 262144 (full IoU row)
static constexpr unsigned SM_MASK = 2048 * 4;       // 8192 (anchor bitmask)
static constexpr unsigned SM_TRU  = 512;
static constexpr unsigned SM_REDV = 128;            // 32 wave slots
static constexpr unsigned SM_REDI = 128;
static constexpr unsigned SM_TOP  = 128;
static constexpr unsigned SMEM_MATCH = SM_ROW + SM_MASK + SM_TRU + SM_REDV + SM_REDI + SM_TOP;
static constexpr unsigned SMEM_HARD  = A_N * 4 + 128 + 128;   // keys + red scratch

// =====================================================================
// Kernel 0: zero the small accumulators (every call: graph replays reuse ws)
// =====================================================================
__global__ void mb_init_kernel(float* acc4, float* loss_pos, int* n_pos,
                               int* n_neg, float* loss_neg) {
  int t = threadIdx.x;
  if (t < 4)  acc4[t] = 0.0f;
  if (t == 0) loss_pos[0] = 0.0f;
  if (t < B_N) { n_pos[t] = 0; n_neg[t] = 0; loss_neg[t] = 0.0f; }
}

// =====================================================================
// Kernel 1: mask-independent precompute, fully parallel. Moves all expf /
// 48-IoU colmax work off the serial matching critical path.
// =====================================================================
__global__ __launch_bounds__(256) void mb_pre_kernel(
    const float* __restrict__ loc,      // [B][A][4]
    const float* __restrict__ priors,   // [A][4]
    const float* __restrict__ targets,  // [B][O][15]
    float* __restrict__ pf_prior,       // [A][4]   point_form(priors)
    float* __restrict__ dec_pf,         // [B][A][4] point_form(decode(loc))
    float* __restrict__ colmax)         // [B][A]
{
  __shared__ float tru[O_N * 4];
  const int blocksPerB = A_N / 256;
  const int b = blockIdx.x / blocksPerB;
  const int a = (blockIdx.x % blocksPerB) * 256 + threadIdx.x;
  if (threadIdx.x < O_N * 4)
    tru[threadIdx.x] = targets[((size_t)b * O_N + (threadIdx.x >> 2)) * 15 + (threadIdx.x & 3)];
  __syncthreads();

  float4 p = reinterpret_cast<const float4*>(priors)[a];
  float px0 = p.x - 0.5f*p.z, py0 = p.y - 0.5f*p.w;
  float px1 = p.x + 0.5f*p.z, py1 = p.y + 0.5f*p.w;
  float4 l = reinterpret_cast<const float4*>(loc)[(size_t)b * A_N + a];
  float cx = p.x + l.x * 0.1f * p.z;
  float cy = p.y + l.y * 0.1f * p.w;
  float w  = p.z * expf(l.z * 0.2f);
  float h  = p.w * expf(l.w * 0.2f);
  float dx0 = cx - 0.5f*w, dy0 = cy - 0.5f*h, dx1 = cx + 0.5f*w, dy1 = cy + 0.5f*h;

  if (b == 0) reinterpret_cast<float4*>(pf_prior)[a] = make_float4(px0,py0,px1,py1);
  reinterpret_cast<float4*>(dec_pf)[(size_t)b * A_N + a] = make_float4(dx0,dy0,dx1,dy1);

  float cm = -1e30f;
  #pragma unroll 4
  for (int i = 0; i < O_N; ++i) {
    float tx0 = tru[i*4+0], ty0 = tru[i*4+1], tx1 = tru[i*4+2], ty1 = tru[i*4+3];
    cm = fmaxf(cm, iou_box(tx0,ty0,tx1,ty1, px0,py0,px1,py1));
    cm = fmaxf(cm, iou_box(tx0,ty0,tx1,ty1, dx0,dy0,dx1,dy1));
  }
  colmax[(size_t)b * A_N + a] = cm;
}

// =====================================================================
// Kernel 2: sequential matching scan. One workgroup (1024 thr) per batch.
// Entire 65536-float IoU row lives in LDS (needs CDNA5's 320KB LDS/WGP).
// =====================================================================
__global__ __launch_bounds__(TPB) void mb_match_kernel(
    const float* __restrict__ pf_prior, // [A][4]
    const float* __restrict__ dec_pf,   // [B][A][4]
    const float* __restrict__ targets,  // [B][O][15]
    unsigned* __restrict__ maskw,       // [B][2048]
    int*      __restrict__ p_idx,       // [B][480]
    int*      __restrict__ p_val)       // [B][480]
{
  extern __shared__ char smem[];
  float*    row  = (float*)smem;
  unsigned* mask = (unsigned*)(smem + SM_ROW);
  float*    tru  = (float*)(smem + SM_ROW + SM_MASK);
  float*    redv = (float*)(smem + SM_ROW + SM_MASK + SM_TRU);
  unsigned* redi = (unsigned*)(smem + SM_ROW + SM_MASK + SM_TRU + SM_REDV);
  float*    topv = (float*)(smem + SM_ROW + SM_MASK + SM_TRU + SM_REDV + SM_REDI);
  unsigned* topi = (unsigned*)((char*)topv + 64);

  const int b = blockIdx.x;
  const int t = threadIdx.x;

  if (t < O_N * 4) tru[t] = targets[((size_t)b * O_N + (t >> 2)) * 15 + (t & 3)];
  for (int w = t; w < 2048; w += TPB) mask[w] = 0u;
  __syncthreads();

  for (int i = 0; i < O_N; ++i) {
    float tx0 = tru[i*4+0], ty0 = tru[i*4+1], tx1 = tru[i*4+2], ty1 = tru[i*4+3];
    for (int phase = 0; phase < 2; ++phase) {
      // ---- fill IoU row: L2-resident float4 box + one IoU per anchor ----
      const float4* boxsrc = (phase == 0)
          ? reinterpret_cast<const float4*>(pf_prior)
          : reinterpret_cast<const float4*>(dec_pf) + (size_t)b * A_N;
      for (int s = 0; s < PER_T; ++s) {
        int a = t + s * TPB;
        float4 bx = boxsrc[a];
        if (s + 1 < PER_T) __builtin_prefetch(&boxsrc[a + TPB], 0, 0);
        row[a] = iou_box(tx0,ty0,tx1,ty1, bx.x,bx.y,bx.z,bx.w);
      }
      __syncthreads();

      // ---- top-10 by repeated masked argmax (ties -> smallest index) ----
      for (int j = 0; j < K_TOP; ++j) {
        float bv = -1e30f; unsigned bi = 0xFFFFFFFFu;
        for (int s = 0; s < PER_T; ++s) {
          int a = t + s * TPB;
          bool m = (mask[a >> 5] >> (a & 31)) & 1u;
          float v = m ? -1.0f : row[a];
          if (v > bv || (v == bv && (unsigned)a < bi)) { bv = v; bi = (unsigned)a; }
        }
        float gv; unsigned gi;
        block_argmax(bv, bi, redv, redi, t, gv, gi);
        if (t == 0) {
          topv[j] = gv; topi[j] = gi;
          mask[gi >> 5] |= (1u << (gi & 31));   // exclude from next extraction
        }
        __syncthreads();
      }

      // ---- validity, fallback, mask cleanup, store selections ----
      if (t == 0) {
        float th = (phase == 0) ? 0.35f : 0.40f;   // P_TH / P_TH+0.05
        int valid[K_TOP]; int anyv = 0;
        for (int j = 0; j < K_TOP; ++j) { valid[j] = topv[j] > th; anyv |= valid[j]; }
        if (phase == 0 && !anyv) valid[0] = 1;     // fallback: force best anchor
        for (int j = 0; j < K_TOP; ++j) {
          if (!valid[j]) { unsigned a = topi[j]; mask[a >> 5] &= ~(1u << (a & 31)); }
          int slot = b * NSEL + i * (2 * K_TOP) + phase * K_TOP + j;
          p_idx[slot] = (int)topi[j];
          p_val[slot] = valid[j];
        }
      }
      __syncthreads();
    }
  }
  for (int w = t; w < 2048; w += TPB) maskw[b * 2048 + w] = mask[w];
}

// =====================================================================
// Kernel 3: per-anchor classification terms. WMMA wave reduction of the
// positive loss; writes masked negative losses for hard mining.
// =====================================================================
__global__ __launch_bounds__(256) void mb_cls_kernel(
    const float* __restrict__ conf,     // [B][A][2]
    const float* __restrict__ colmax,   // [B][A]
    const unsigned* __restrict__ maskw, // [B][2048]
    float* __restrict__ lcneg,          // [B][A]
    float* __restrict__ loss_pos,
    int* __restrict__ n_pos, int* __restrict__ n_neg)
{
  __shared__ float wacc[8];
  __shared__ int sPos, sNeg;
  if (threadIdx.x == 0) { sPos = 0; sNeg = 0; }
  __syncthreads();

  const int idx = blockIdx.x * 256 + threadIdx.x;   // over B*A
  const int b = idx >> 16;
  const int a = idx & (A_N - 1);

  float2 c = reinterpret_cast<const float2*>(conf)[idx];
  float m = fmaxf(c.x, c.y);
  float lse = m + logf(expf(c.x - m) + expf(c.y - m));

  bool pos = (maskw[b * 2048 + (a >> 5)] >> (a & 31)) & 1u;
  bool neg = (!pos) && (colmax[idx] < 0.4f);        // N_TH; masked cols excluded
  lcneg[idx] = neg ? (lse - c.x) : (-__builtin_huge_valf());

  float posterm = pos ? (lse - c.y) : 0.0f;
  float wsum = wave_sum32(posterm);                 // v_wmma_f32_16x16x4_f32
  int lane = threadIdx.x & 31, wid = threadIdx.x >> 5;
  if (lane == 0) wacc[wid] = wsum;
  atomicAdd(&sPos, pos ? 1 : 0);
  atomicAdd(&sNeg, neg ? 1 : 0);
  __syncthreads();
  if (threadIdx.x == 0) {
    float ssum = wacc[0]+wacc[1]+wacc[2]+wacc[3]+wacc[4]+wacc[5]+wacc[6]+wacc[7];
    atomicAdd(loss_pos, ssum);
    atomicAdd(&n_pos[b], sPos);
    atomicAdd(&n_neg[b], sNeg);
  }
}

// =====================================================================
// Kernel 4: exact hard-negative mining. The 256KB row is pulled into LDS
// by the Tensor Data Mover (tensor_load_to_lds, one issue from wave 0,
// TENSORcnt-synchronized), then a 32-step radix binary search finds the
// exact n-th largest value; sum(top-n) = sum(key>thr) + thr*(n - cnt_gt).
// =====================================================================
__global__ __launch_bounds__(TPB) void mb_hardneg_kernel(
    const float* __restrict__ lcneg, const int* __restrict__ n_pos,
    const int* __restrict__ n_neg, float* __restrict__ loss_neg)
{
  extern __shared__ char smem[];
  unsigned* keys = (unsigned*)smem;                   // 65536 words
  int*      redc = (int*)(smem + A_N * 4);            // 32 slots
  float*    redf = (float*)(smem + A_N * 4 + 128);    // 32 slots

  const int b = blockIdx.x;
  const int t = threadIdx.x;

  // ---- TDM: DMA 65536 floats (2D tile 256x256, 4B elems) global -> LDS ----
  // TDM ignores EXEC, so gate the issue on a *scalar* wave-id branch.
  int wid_u = __builtin_amdgcn_readfirstlane(threadIdx.x >> 5);
  if (wid_u == 0) {
    unsigned long long gaddr =
        (unsigned long long)(uintptr_t)(lcneg + (size_t)b * A_N);
    unsigned ldsoff = (unsigned)(uintptr_t)(void*)keys;
    // D# group0: count=1 | lds_addr | global_addr[56:0] | type=2
    v4u g0 = { 1u, ldsoff, (unsigned)gaddr,
               (unsigned)((gaddr >> 32) & 0x01FFFFFFu) | 0x80000000u };
    // D# group1: data_size=4B(2); tensor_dim0=256; tensor_dim1=256;
    // tile_dim0=256; tile_dim1=256; tensor_dim0_stride=256
    v8i g1 = { 0x20000, 0x01000000, 0x01000000, 0x01000000,
               0x100, 0x100, 0, 0 };
    v4i z4 = { 0, 0, 0, 0 };
#if defined(__clang_major__) && (__clang_major__ >= 23)
    v8i z8 = { 0, 0, 0, 0, 0, 0, 0, 0 };
    __builtin_amdgcn_tensor_load_to_lds(g0, g1, z4, z4, z8, 0);
#else
    __builtin_amdgcn_tensor_load_to_lds(g0, g1, z4, z4, 0);
#endif
    __builtin_amdgcn_s_wait_tensorcnt(0);
  }
  __syncthreads();

  // in-place float-bits -> monotone sortable key
  for (int s = 0; s < PER_T; ++s) {
    int a = t + s * TPB;
    keys[a] = bits2key(keys[a]);
  }
  __syncthreads();

  int n = n_pos[b] * 7;                               // NEGPOS
  int nn = n_neg[b];
  if (n > nn) n = nn;
  if (n > 0) {
    unsigned thr = 0u;
    for (int bit = 31; bit >= 0; --bit) {
      unsigned cand = thr | (1u << bit);
      int c = 0;
      for (int s = 0; s < PER_T; ++s) c += (keys[t + s * TPB] >= cand) ? 1 : 0;
      int cnt = block_sum_i(c, redc, t);
      if (cnt >= n) thr = cand;                       // uniform across block
    }
    int cg = 0; float sg = 0.0f;
    for (int s = 0; s < PER_T; ++s) {
      unsigned k = keys[t + s * TPB];
      if (k > thr) { cg++; sg += key2f(k); }
    }
    int cnt_gt = block_sum_i(cg, redc, t);
    float sum_gt = block_sum_f(sg, redf, t);
    if (t == 0) loss_neg[b] = sum_gt + key2f(thr) * (float)(n - cnt_gt);
  } // n==0 -> loss_neg[b] stays 0 from init kernel
}

// =====================================================================
// Kernel 5: localization + landmark losses over the 480 selections/batch
// =====================================================================
__global__ __launch_bounds__(NSEL) void mb_locloss_kernel(
    const float* __restrict__ loc,      // [B][A][4]
    const float* __restrict__ landm,    // [B][A][6]
    const float* __restrict__ priors,   // [A][4]
    const float* __restrict__ targets,  // [B][O][15]
    const int* __restrict__ p_idx, const int* __restrict__ p_val,
    float* __restrict__ acc4)           // [loss_l, N, loss_landm, N1]
{
  const int b = blockIdx.x;
  const int s = threadIdx.x;            // 0..479
  const int slot = b * NSEL + s;
  if (!p_val[slot]) return;
  const int i = s / (2 * K_TOP);
  const int a = p_idx[slot];
  const float* T = &targets[((size_t)b * O_N + i) * 15];
  float4 pr = reinterpret_cast<const float4*>(priors)[a];

  // encode(truth, prior) vs loc_data
  float gcx = ((T[0] + T[2]) * 0.5f - pr.x) / (0.1f * pr.z);
  float gcy = ((T[1] + T[3]) * 0.5f - pr.y) / (0.1f * pr.w);
  float gw  = logf((T[2] - T[0]) / pr.z) / 0.2f;
  float gh  = logf((T[3] - T[1]) / pr.w) / 0.2f;
  float4 L = reinterpret_cast<const float4*>(loc)[(size_t)b * A_N + a];
  float ll = smooth_l1(L.x - gcx) + smooth_l1(L.y - gcy) +
             smooth_l1(L.z - gw)  + smooth_l1(L.w - gh);
  atomicAdd(&acc4[0], ll);
  atomicAdd(&acc4[1], 1.0f);

  if (T[14] == 1.0f) {                  // label gate (lvalid)
    const float* LM = &landm[((size_t)b * A_N + a) * 6];
    float sl = 0.0f;
    #pragma unroll
    for (int k = 0; k < 3; ++k) {       // encode_landm(...)[..., :6]
      float gx = (T[4 + 2*k] - pr.x) / (0.1f * pr.z);
      float gy = (T[5 + 2*k] - pr.y) / (0.1f * pr.w);
      sl += smooth_l1(LM[2*k] - gx) + smooth_l1(LM[2*k + 1] - gy);
    }
    atomicAdd(&acc4[2], sl);
    atomicAdd(&acc4[3], 1.0f);
  }
}

// =====================================================================
// Kernel 6: final combine (WMMA wave-sum of the 16 per-batch neg losses)
// =====================================================================
__global__ __launch_bounds__(32) void mb_final_kernel(
    const float* __restrict__ acc4, const float* __restrict__ loss_pos,
    const float* __restrict__ loss_neg, float* __restrict__ out)
{
  int t = threadIdx.x;
  float v = (t < B_N) ? loss_neg[t] : 0.0f;
  float negsum = wave_sum32(v);
  if (t == 0) {
    out[0] = acc4[0] / acc4[1];                      // loss_l / N
    out[1] = (loss_pos[0] + negsum) / acc4[1];       // loss_cls / N
    out[2] = acc4[2] / acc4[3];                      // loss_landm / N1
  }
}

// =====================================================================
extern "C" void kernel_launch(void* const* d_in, const int* in_sizes, int n_in,
                              void* d_out, int out_size, void* d_ws, size_t ws_size,
                              hipStream_t stream) {
  const float* loc     = (const float*)d_in[0];   // 16*65536*4
  const float* conf    = (const float*)d_in[1];   // 16*65536*2
  const float* landm   = (const float*)d_in[2];   // 16*65536*6
  const float* priors  = (const float*)d_in[3];   // 65536*4
  const float* targets = (const float*)d_in[4];   // 16*24*15
  float* out = (float*)d_out;

  // ---- workspace carve (~25.2 MB) ----
  char* ws = (char*)d_ws;
  size_t off = 0;
  float*    colmax   = (float*)(ws + off); off += (size_t)B_N * A_N * 4;       // 4MB
  float*    lcneg    = (float*)(ws + off); off += (size_t)B_N * A_N * 4;       // 4MB
  float*    dec_pf   = (float*)(ws + off); off += (size_t)B_N * A_N * 16;      // 16MB
  float*    pf_prior = (float*)(ws + off); off += (size_t)A_N * 16;            // 1MB
  unsigned* maskw    = (unsigned*)(ws + off); off += (size_t)B_N * 2048 * 4;   // 128KB
  int*      p_idx    = (int*)(ws + off); off += (size_t)B_N * NSEL * 4;
  int*      p_val    = (int*)(ws + off); off += (size_t)B_N * NSEL * 4;
  float*    acc4     = (float*)(ws + off); off += 16;   // loss_l, N, loss_landm, N1
  float*    loss_pos = (float*)(ws + off); off += 16;
  int*      n_pos    = (int*)(ws + off); off += 64;
  int*      n_neg    = (int*)(ws + off); off += 64;
  float*    loss_neg = (float*)(ws + off); off += 64;

  // allow >64KB dynamic LDS (CDNA5: 320KB/WGP)
  (void)hipFuncSetAttribute((const void*)mb_match_kernel,
      hipFuncAttributeMaxDynamicSharedMemorySize, SMEM_MATCH);
  (void)hipFuncSetAttribute((const void*)mb_hardneg_kernel,
      hipFuncAttributeMaxDynamicSharedMemorySize, SMEM_HARD);

  mb_init_kernel<<<1, 64, 0, stream>>>(acc4, loss_pos, n_pos, n_neg, loss_neg);
  mb_pre_kernel<<<(B_N * A_N) / 256, 256, 0, stream>>>(
      loc, priors, targets, pf_prior, dec_pf, colmax);
  mb_match_kernel<<<B_N, TPB, SMEM_MATCH, stream>>>(
      pf_prior, dec_pf, targets, maskw, p_idx, p_val);
  mb_cls_kernel<<<(B_N * A_N) / 256, 256, 0, stream>>>(
      conf, colmax, maskw, lcneg, loss_pos, n_pos, n_neg);
  mb_hardneg_kernel<<<B_N, TPB, SMEM_HARD, stream>>>(
      lcneg, n_pos, n_neg, loss_neg);
  mb_locloss_kernel<<<B_N, NSEL, 0, stream>>>(
      loc, landm, priors, targets, p_idx, p_val, acc4);
  mb_final_kernel<<<1, 32, 0, stream>>>(acc4, loss_pos, loss_neg, out);
}